// Attn_44281112822483
// MI455X (gfx1250) — compile-verified
//
#include <hip/hip_runtime.h>
#include <math.h>

// Problem constants (from reference): S=4096, B=64, H=300
#define SEQ 4096
#define BAT 64
#define HID 300
#define HIDP 304   // padded H for the v workspace (multiple of 16)

typedef __attribute__((ext_vector_type(2))) float v2f;
typedef __attribute__((ext_vector_type(8))) float v8f;

// ---------------------------------------------------------------------------
// Kernel 1: v[b,n] = sum_o hidden[b,o] * W[o,n]   (M=64, N=300->304, K=300)
// One wave per 16x16 C tile using V_WMMA_F32_16X16X4_F32 (fp32-exact path).
// Grid: 19 blocks (N tiles) x 128 threads (4 waves = 4 M tiles).
// Branchless edge handling: clamp column address, v_cndmask the value to 0.
// ---------------------------------------------------------------------------
__global__ void __launch_bounds__(128)
vproj_wmma_kernel(const float* __restrict__ hidden,
                  const float* __restrict__ W,
                  float* __restrict__ v_ws) {
    const int lane = threadIdx.x & 31;
    const int wave = threadIdx.x >> 5;      // 0..3 -> M tile
    const int m0   = wave * 16;
    const int n0   = blockIdx.x * 16;       // 0..18
    const int lid  = lane & 15;
    const int half = lane >> 4;             // 0: K0/K1, 1: K2/K3
    const int n    = n0 + lid;
    const bool nok = (n < HID);
    const int nc   = nok ? n : (HID - 1);   // clamped, always in-bounds

    v8f acc = {};
    for (int k = 0; k < HID; k += 4) {
        const int kk = k + 2 * half;
        // A fragment: lane(0-15)=M row, v0=A[m,kk], v1=A[m,kk+1]  (b64 load)
        const v2f a = *(const v2f*)(hidden + (m0 + lid) * HID + kk);
        // B fragment: lane(0-15)=N col, v0=B[kk,n], v1=B[kk+1,n]
        float bx = W[kk * HID + nc];
        float by = W[(kk + 1) * HID + nc];
        v2f bb;
        bb.x = nok ? bx : 0.0f;             // v_cndmask, no exec diamond
        bb.y = nok ? by : 0.0f;
        // 8 args: (neg_a, A, neg_b, B, c_mod, C, reuse_a, reuse_b)
        acc = __builtin_amdgcn_wmma_f32_16x16x4_f32(
            false, a, false, bb, (short)0, acc, false, false);
    }
    // C/D layout: VGPR r -> row (r + 8*half), col = n0+lid. Padded store.
    #pragma unroll
    for (int r = 0; r < 8; ++r) {
        v_ws[(m0 + r + 8 * half) * HIDP + n0 + lid] = acc[r];
    }
}

// ---------------------------------------------------------------------------
// Kernel 1b: c[b] = hidden[b] . attn_b   (trivial)
// ---------------------------------------------------------------------------
__global__ void __launch_bounds__(64)
bias_dot_kernel(const float* __restrict__ hidden,
                const float* __restrict__ bias,
                float* __restrict__ c_ws) {
    const int b = threadIdx.x;
    float acc = 0.0f;
    for (int o = 0; o < HID; ++o) acc += hidden[b * HID + o] * bias[o];
    c_ws[b] = acc;
}

// ---------------------------------------------------------------------------
// Kernel 2 (the 300 MB stream, bandwidth-bound): one wave per (s,b) row.
// energies[b,s] = v[b] . enc[s,b,:] + c[b]
// Consecutive waves hit consecutive 1200 B rows -> fully coalesced b128 loads.
// Fully unrolled, branch-free: 75 float4s = 32 + 32 + 11(masked, clamped).
// ---------------------------------------------------------------------------
__global__ void __launch_bounds__(256)
energies_kernel(const float* __restrict__ enc,
                const float* __restrict__ v_ws,
                const float* __restrict__ c_ws,
                float* __restrict__ e_ws) {
    const int lane = threadIdx.x & 31;
    const int wave = threadIdx.x >> 5;
    const int widx = blockIdx.x * 8 + wave;   // 0 .. S*B-1
    const int s = widx >> 6;                  // / BAT
    const int b = widx & (BAT - 1);

    const float4* __restrict__ erow = (const float4*)(enc + (size_t)(s * BAT + b) * HID);
    const float4* __restrict__ vrow = (const float4*)(v_ws + b * HIDP);

    // chunk 0: h4 = lane        (0..31  < 75, always valid)
    const float4 e0 = erow[lane];
    const float4 w0 = vrow[lane];
    // chunk 1: h4 = lane + 32   (32..63 < 75, always valid)
    const float4 e1 = erow[lane + 32];
    const float4 w1 = vrow[lane + 32];
    // chunk 2: h4 = lane + 64   (valid for lanes 0..10); clamp + mask
    const int   h2  = lane + 64;
    const int   h2c = (h2 < 75) ? h2 : 74;
    const float msk = (h2 < 75) ? 1.0f : 0.0f;
    const float4 e2 = erow[h2c];
    const float4 w2 = vrow[h2c];

    float acc = e0.x * w0.x + e0.y * w0.y + e0.z * w0.z + e0.w * w0.w
              + e1.x * w1.x + e1.y * w1.y + e1.z * w1.z + e1.w * w1.w
              + msk * (e2.x * w2.x + e2.y * w2.y + e2.z * w2.z + e2.w * w2.w);

    // wave32 butterfly reduction
    #pragma unroll
    for (int off = 16; off > 0; off >>= 1) acc += __shfl_xor(acc, off, 32);
    if (lane == 0) e_ws[b * SEQ + s] = acc + c_ws[b];
}

// ---------------------------------------------------------------------------
// Kernel 3: softmax over s (4096) per batch row; output [1,B,S] = b*S+s.
// ---------------------------------------------------------------------------
__global__ void __launch_bounds__(256)
softmax_kernel(const float* __restrict__ e_ws, float* __restrict__ out) {
    __shared__ float red[256];
    const int b = blockIdx.x;
    const int t = threadIdx.x;
    const float* row = e_ws + b * SEQ;

    float m = -INFINITY;
    for (int s = t; s < SEQ; s += 256) m = fmaxf(m, row[s]);
    red[t] = m; __syncthreads();
    for (int w = 128; w > 0; w >>= 1) {
        if (t < w) red[t] = fmaxf(red[t], red[t + w]);
        __syncthreads();
    }
    m = red[0]; __syncthreads();

    float sum = 0.0f;
    for (int s = t; s < SEQ; s += 256) sum += __expf(row[s] - m);
    red[t] = sum; __syncthreads();
    for (int w = 128; w > 0; w >>= 1) {
        if (t < w) red[t] += red[t + w];
        __syncthreads();
    }
    const float inv = 1.0f / red[0];
    for (int s = t; s < SEQ; s += 256) out[b * SEQ + s] = __expf(row[s] - m) * inv;
}

// ---------------------------------------------------------------------------
extern "C" void kernel_launch(void* const* d_in, const int* in_sizes, int n_in,
                              void* d_out, int out_size, void* d_ws, size_t ws_size,
                              hipStream_t stream) {
    (void)in_sizes; (void)n_in; (void)out_size; (void)ws_size;
    const float* hidden = (const float*)d_in[0];   // [1,64,300]
    const float* enc    = (const float*)d_in[1];   // [4096,64,300]
    // d_in[2] = question_vector: unused by the reference computation
    const float* attn_W = (const float*)d_in[3];   // [300,300]
    const float* attn_b = (const float*)d_in[4];   // [300]
    float* out = (float*)d_out;                    // [1,64,4096]

    // Workspace layout (floats): v[64*304] | c[64] | energies[64*4096]
    float* v_ws = (float*)d_ws;
    float* c_ws = v_ws + BAT * HIDP;               // +19456
    float* e_ws = c_ws + BAT;                      // +19520 (16B aligned)

    vproj_wmma_kernel<<<(HIDP / 16), 128, 0, stream>>>(hidden, attn_W, v_ws);
    bias_dot_kernel<<<1, 64, 0, stream>>>(hidden, attn_b, c_ws);
    energies_kernel<<<(SEQ * BAT) / 8, 256, 0, stream>>>(enc, v_ws, c_ws, e_ws);
    softmax_kernel<<<BAT, 256, 0, stream>>>(e_ws, out);
}